// SelfSupervisedLoss_55654186221903
// MI455X (gfx1250) — compile-verified
//
#include <hip/hip_runtime.h>

typedef __attribute__((ext_vector_type(2))) float v2f;
typedef __attribute__((ext_vector_type(8))) float v8f;

#define NB_   4
#define NC_   64
#define NH_   200
#define NW_   200
#define HW_   (NH_ * NW_)            // 40000
#define PIX_PER_I (NB_ * HW_)        // 160000
#define TOTAL_PIX (2 * PIX_PER_I)    // 320000
#define BLOCK 256
#define NBLOCKS (TOTAL_PIX / BLOCK)  // 1250 exactly, no tail

// Exact f32 sum of one value per lane across the wave32, via
// V_WMMA_F32_16X16X4_F32 with an all-ones B matrix.
// With B == 1 everywhere, D[m][n] = sum_k A[m][k]; summing a full column of D
// equals the sum of ALL elements of A (layout-robust), i.e. the 32 lane values.
// C/D layout: column n lives in lanes n and n+16 (8 VGPRs each).
__device__ __forceinline__ float wave_sum32(float v) {
    v2f a; a[0] = v;    a[1] = 0.0f;   // 32 lanes x 2 regs = all 64 A slots; 32 are v, 32 are 0
    v2f b; b[0] = 1.0f; b[1] = 1.0f;   // B(4x16) all ones
    v8f c = {};
    c = __builtin_amdgcn_wmma_f32_16x16x4_f32(
        /*neg_a=*/false, a, /*neg_b=*/false, b,
        /*c_mod=*/(short)0, c, /*reuse_a=*/false, /*reuse_b=*/false);
    float s = ((c[0] + c[1]) + (c[2] + c[3])) + ((c[4] + c[5]) + (c[6] + c[7]));
    return __shfl(s, 0, 32) + __shfl(s, 16, 32); // column 0: lanes 0 (M=0..7) + 16 (M=8..15)
}

__global__ __launch_bounds__(BLOCK)
void warp_mse_kernel(const float* __restrict__ feat,   // (4,3,64,200,200)
                     const float* __restrict__ flow,   // (4,3,2,200,200)
                     float* __restrict__ blk) {
    const int t = blockIdx.x * BLOCK + threadIdx.x;
    const int w = t % NW_;
    const int h = (t / NW_) % NH_;
    const int n = (t / HW_) % NB_;
    const int i = (t / PIX_PER_I) + 1;   // i in {1,2}

    // flow channels x,y for frame i
    const size_t fbase = (((size_t)n * 3 + i) * 2) * HW_ + (size_t)h * NW_ + w;
    const float fx = flow[fbase];
    const float fy = flow[fbase + HW_];

    // grid_sample(align_corners=False) coordinate math, same order as reference
    const float gx = (-1.0f + 2.0f * (float)w / (float)(NW_ - 1)) + fx / ((NW_ - 1) * 0.5f);
    const float gy = (-1.0f + 2.0f * (float)h / (float)(NH_ - 1)) + fy / ((NH_ - 1) * 0.5f);
    const float x = ((gx + 1.0f) * (float)NW_ - 1.0f) * 0.5f;
    const float y = ((gy + 1.0f) * (float)NH_ - 1.0f) * 0.5f;

    const float x0f = floorf(x), y0f = floorf(y);
    const float wx1 = x - x0f,  wy1 = y - y0f;
    const float wx0 = 1.0f - wx1, wy0 = 1.0f - wy1;
    const int ix0 = (int)x0f, iy0 = (int)y0f;
    const int ix1 = ix0 + 1,  iy1 = iy0 + 1;

    const bool vx0 = (ix0 >= 0) && (ix0 < NW_);
    const bool vx1 = (ix1 >= 0) && (ix1 < NW_);
    const bool vy0 = (iy0 >= 0) && (iy0 < NH_);
    const bool vy1 = (iy1 >= 0) && (iy1 < NH_);
    const int cx0 = min(max(ix0, 0), NW_ - 1), cx1 = min(max(ix1, 0), NW_ - 1);
    const int cy0 = min(max(iy0, 0), NH_ - 1), cy1 = min(max(iy1, 0), NH_ - 1);

    // validity folded into weights (matches reference: v * valid then * weight)
    float w00 = wx0 * wy0 * ((vx0 && vy0) ? 1.0f : 0.0f);
    float w10 = wx1 * wy0 * ((vx1 && vy0) ? 1.0f : 0.0f);
    float w01 = wx0 * wy1 * ((vx0 && vy1) ? 1.0f : 0.0f);
    float w11 = wx1 * wy1 * ((vx1 && vy1) ? 1.0f : 0.0f);

    // ones-channel interpolation == sum of valid weights; mask = (>0.999)
    const float wsum = (w00 + w10) + (w01 + w11);
    const float m = (wsum > 0.999f) ? 1.0f : 0.0f;
    w00 *= m; w10 *= m; w01 *= m; w11 *= m;   // mask folded in -> branch-free channel loop

    const float* __restrict__ cur = feat + (((size_t)n * 3 + i)       * NC_) * HW_;
    const float* __restrict__ prv = feat + (((size_t)n * 3 + (i - 1)) * NC_) * HW_;
    size_t o00 = (size_t)cy0 * NW_ + cx0;
    size_t o10 = (size_t)cy0 * NW_ + cx1;
    size_t o01 = (size_t)cy1 * NW_ + cx0;
    size_t o11 = (size_t)cy1 * NW_ + cx1;
    size_t op  = (size_t)h * NW_ + w;

    float acc = 0.0f;
#pragma unroll 8
    for (int c = 0; c < NC_; ++c) {
        const float wv = w00 * cur[o00] + w10 * cur[o10]
                       + w01 * cur[o01] + w11 * cur[o11];
        const float d = prv[op] - wv;
        acc = fmaf(d, d, acc);
        o00 += HW_; o10 += HW_; o01 += HW_; o11 += HW_; op += HW_;
    }

    // wave-level exact sum via WMMA, then LDS combine of the 8 waves
    const float wave_total = wave_sum32(acc);
    __shared__ float sm[BLOCK / 32];
    const int lane = threadIdx.x & 31;
    const int wid  = threadIdx.x >> 5;
    if (lane == 0) sm[wid] = wave_total;
    __syncthreads();
    if (threadIdx.x == 0) {
        float tsum = 0.0f;
        for (int j = 0; j < BLOCK / 32; ++j) tsum += sm[j];
        blk[blockIdx.x] = tsum;
    }
}

// Single-wave deterministic finalize: strided serial accumulation (fixed order)
// of the 1250 block sums, then WMMA wave reduction, then scale by 1/(N*C*H*W).
__global__ __launch_bounds__(32)
void finalize_kernel(const float* __restrict__ blk, float* __restrict__ out) {
    const int lane = threadIdx.x;
    float s = 0.0f;
    for (int k = lane; k < NBLOCKS; k += 32) s += blk[k];
    const float total = wave_sum32(s);
    if (lane == 0) out[0] = total * (1.0f / (float)((size_t)NB_ * NC_ * HW_));
}

extern "C" void kernel_launch(void* const* d_in, const int* in_sizes, int n_in,
                              void* d_out, int out_size, void* d_ws, size_t ws_size,
                              hipStream_t stream) {
    const float* feat = (const float*)d_in[0];  // (4,3,64,200,200) f32
    const float* flow = (const float*)d_in[1];  // (4,3,2,200,200) f32
    float* blk = (float*)d_ws;                  // 1250 f32 partials, fully overwritten each call

    warp_mse_kernel<<<NBLOCKS, BLOCK, 0, stream>>>(feat, flow, blk);
    finalize_kernel<<<1, 32, 0, stream>>>(blk, (float*)d_out);
}